// XAMIRegionProposalNetwork_8117488190272
// MI455X (gfx1250) — compile-verified
//
#include <hip/hip_runtime.h>
#include <stdint.h>

// ---------------- problem constants (match reference) ----------------
#define A_TOTAL  159882
#define NIMG     8
#define KTOT     4507          // 1000+1000+1000+1000+507 selected per image
#define KPAD     4512          // KTOT padded to multiple of 32
#define NW       141           // KPAD / 32 mask words per row
#define ROWS_PER_BLK 8         // row tiles per 256-thread mask block (8 waves)
#define NRB      18            // ceil(NW / ROWS_PER_BLK)
#define NLEV     5
#define POST_N   1000
#define NSORT    8192          // power of two >= KTOT for bitonic sort
#define IMG_WH   800.0f
#define NMS_T    0.7f
#define MIN_SZ   0.001f
#define CLIP_DW  4.135166556742356f   // log(1000/16)
#define LVL_OFFS 2048.0f

__device__ const int c_lvl_n[NLEV]   = {120000, 30000, 7500, 1875, 507};
__device__ const int c_lvl_off[NLEV] = {0, 120000, 150000, 157500, 159375};
__device__ const int c_out_off[NLEV] = {0, 1000, 2000, 3000, 4000};
__device__ const int c_lvl_k[NLEV]   = {1000, 1000, 1000, 1000, 507};

// monotonic float->uint mapping: ascending uint <=> ascending float
__device__ __forceinline__ unsigned mkey(float x) {
  unsigned u = __float_as_uint(x);
  return u ^ ((u >> 31) ? 0xFFFFFFFFu : 0x80000000u);
}

// ---------------- K0: zero the per-image invalid bitmask ----------------
__global__ void init_kernel(unsigned* __restrict__ init_removed) {
  int g = blockIdx.x * blockDim.x + threadIdx.x;
  if (g < NIMG * NW) init_removed[g] = 0u;
}

// ---------------- K1: exact top-k per (image, level) via 4-pass radix select
__global__ __launch_bounds__(1024) void topk_kernel(const float* __restrict__ obj,
                                                    float* __restrict__ cand_score,
                                                    int* __restrict__ cand_aidx) {
  const int lvl = blockIdx.x, img = blockIdx.y, tid = threadIdx.x;
  const int n = c_lvl_n[lvl], off = c_lvl_off[lvl], k = c_lvl_k[lvl];
  const float* p = obj + (size_t)img * A_TOTAL + off;

  __shared__ int hist[256];
  __shared__ unsigned s_prefix;
  __shared__ int s_kk, s_gt, s_eq;

  unsigned prefix = 0;
  int kk = k;                       // remaining slots at current prefix
  for (int pass = 0; pass < 4; ++pass) {
    const int shift = 24 - 8 * pass;
    for (int b = tid; b < 256; b += 1024) hist[b] = 0;
    __syncthreads();
    for (int i = tid; i < n; i += 1024) {
      __builtin_prefetch(p + i + 8192, 0, 0);     // global_prefetch_b8
      unsigned u = mkey(p[i]);
      if (pass == 0 || (u >> (shift + 8)) == prefix)
        atomicAdd(&hist[(u >> shift) & 255], 1);
    }
    __syncthreads();
    if (tid == 0) {
      int cum = 0, b = 255;
      for (; b > 0; --b) { int c = hist[b]; if (cum + c >= kk) break; cum += c; }
      s_kk = kk - cum;
      s_prefix = (prefix << 8) | (unsigned)b;
    }
    __syncthreads();
    prefix = s_prefix; kk = s_kk;
    __syncthreads();                // protect hist before next-pass zeroing
  }
  const unsigned T = prefix;        // exact 32-bit threshold key
  if (tid == 0) { s_gt = 0; s_eq = 0; }
  __syncthreads();
  for (int i = tid; i < n; i += 1024) {
    float x = p[i];
    unsigned u = mkey(x);
    int slot = -1;
    if (u > T)       slot = atomicAdd(&s_gt, 1);                 // exactly k-kk of these
    else if (u == T) { int e = atomicAdd(&s_eq, 1); if (e < kk) slot = (k - kk) + e; }
    if (slot >= 0) {
      size_t g = (size_t)img * KTOT + c_out_off[lvl] + slot;
      cand_score[g] = x;            // keep logit; sigmoid is monotonic
      cand_aidx[g]  = off + i;
    }
  }
}

// ---------------- K2: decode + clip + validity for the 8*4507 candidates ----
__global__ void decode_kernel(const float* __restrict__ deltas,
                              const float* __restrict__ anchors,
                              const int* __restrict__ cand_aidx,
                              float4* __restrict__ cand_box,
                              float4* __restrict__ cand_obox,
                              int* __restrict__ cand_valid) {
  int g = blockIdx.x * blockDim.x + threadIdx.x;
  if (g >= NIMG * KTOT) return;
  int img = g / KTOT;
  int a = cand_aidx[g];
  const float* an = anchors + (size_t)a * 4;
  const float* de = deltas + ((size_t)img * A_TOTAL + a) * 4;
  float w = an[2] - an[0], h = an[3] - an[1];
  float cx = an[0] + 0.5f * w, cy = an[1] + 0.5f * h;
  float dx = de[0], dy = de[1];
  float dw = fminf(de[2], CLIP_DW), dh = fminf(de[3], CLIP_DW);
  float pcx = dx * w + cx, pcy = dy * h + cy;
  float pw = expf(dw) * w, ph = expf(dh) * h;
  float x1 = pcx - 0.5f * pw, y1 = pcy - 0.5f * ph;
  float x2 = pcx + 0.5f * pw, y2 = pcy + 0.5f * ph;
  x1 = fminf(fmaxf(x1, 0.f), IMG_WH); y1 = fminf(fmaxf(y1, 0.f), IMG_WH);
  x2 = fminf(fmaxf(x2, 0.f), IMG_WH); y2 = fminf(fmaxf(y2, 0.f), IMG_WH);
  int lvl = (a < 120000) ? 0 : (a < 150000) ? 1 : (a < 157500) ? 2 : (a < 159375) ? 3 : 4;
  float o = (float)lvl * LVL_OFFS;  // batched_nms per-level offset
  cand_box[g]  = make_float4(x1, y1, x2, y2);
  cand_obox[g] = make_float4(x1 + o, y1 + o, x2 + o, y2 + o);
  cand_valid[g] = (x2 - x1 >= MIN_SZ) && (y2 - y1 >= MIN_SZ);
}

// ---------------- K3: per-image bitonic sort (descending score, invalid last)
__global__ __launch_bounds__(1024) void sort_kernel(const float* __restrict__ cand_score,
                                                    const int* __restrict__ cand_valid,
                                                    int* __restrict__ perm) {
  const int img = blockIdx.x, tid = threadIdx.x;
  __shared__ unsigned long long keys[NSORT];   // 64KB of the 320KB WGP LDS
  for (int i = tid; i < NSORT; i += 1024) {
    unsigned long long kv;
    if (i < KTOT) {
      size_t g = (size_t)img * KTOT + i;
      unsigned u = mkey(cand_score[g]);
      unsigned hi = cand_valid[g] ? ~u : 0xFFFFFFFFu;   // score desc; invalid -> last
      kv = ((unsigned long long)hi << 32) | (unsigned)i;
    } else kv = ~0ull;                                   // padding sorts last
    keys[i] = kv;
  }
  __syncthreads();
  for (unsigned sz = 2; sz <= NSORT; sz <<= 1) {
    for (unsigned st = sz >> 1; st; st >>= 1) {
      for (int i = tid; i < NSORT; i += 1024) {
        int j = i ^ (int)st;
        if (j > i) {
          bool asc = ((i & sz) == 0);
          unsigned long long a = keys[i], b = keys[j];
          if ((a > b) == asc) { keys[i] = b; keys[j] = a; }
        }
      }
      __syncthreads();
    }
  }
  for (int s = tid; s < KTOT; s += 1024)
    perm[(size_t)img * KTOT + s] = (int)(keys[s] & 0xFFFFFFFFu);
}

// ---------------- K4: gather into sorted order; mark invalid entries --------
__global__ void gather_kernel(const int* __restrict__ perm,
                              const float4* __restrict__ cand_box,
                              const float4* __restrict__ cand_obox,
                              const float* __restrict__ cand_score,
                              const int* __restrict__ cand_valid,
                              float4* __restrict__ sbox,
                              float4* __restrict__ sobox,
                              float* __restrict__ sscore,
                              unsigned* __restrict__ init_removed) {
  int g = blockIdx.x * blockDim.x + threadIdx.x;
  if (g >= NIMG * KPAD) return;
  int img = g / KPAD, s = g % KPAD;
  if (s < KTOT) {
    size_t cp = (size_t)img * KTOT + perm[(size_t)img * KTOT + s];
    sbox[g] = cand_box[cp];
    sobox[g] = cand_obox[cp];
    sscore[g] = cand_score[cp];
    if (!cand_valid[cp])
      atomicOr(&init_removed[img * NW + (s >> 5)], 1u << (s & 31));
  } else {                      // padding: far-away zero-area box, IoU == 0 vs real boxes
    sbox[g] = make_float4(0.f, 0.f, 0.f, 0.f);
    sobox[g] = make_float4(1e8f, 1e8f, 1e8f, 1e8f);
    sscore[g] = 0.f;
  }
}

// ---------------- K5: IoU suppression-mask tiles -----------------------------
// 256 threads = 8 wave32s per block. Wave 0 stages the 32 column boxes into LDS
// once via the gfx1250 async global->LDS DMA; each wave then masks a different
// 32-row tile against the shared column tile (8x reuse, 8x fewer blocks).
__global__ __launch_bounds__(256) void nms_mask_kernel(const float4* __restrict__ sobox,
                                                       unsigned* __restrict__ mask) {
  const int colB = blockIdx.y, img = blockIdx.z;
  const int lane = threadIdx.x & 31;
  const int wave = threadIdx.x >> 5;
  const int rowB = blockIdx.x * ROWS_PER_BLK + wave;

  __shared__ float4 cb[32];                            // sole LDS object -> offset 0
  if (threadIdx.x < 32) {
    // one lane per 16B box: wave32 == tile width
    const float4* gp = sobox + (size_t)img * KPAD + colB * 32 + lane;
    unsigned lds_off = (unsigned)(lane * sizeof(float4));
    unsigned long long ga = (unsigned long long)(uintptr_t)gp;
    asm volatile("global_load_async_to_lds_b128 %0, %1, off\n\t"
                 "s_wait_asynccnt 0"
                 :: "v"(lds_off), "v"(ga)
                 : "memory");
  }
  __syncthreads();                                     // publish cb to all 8 waves

  if (rowB >= NW) return;                              // tail waves (141 % 8 != 0)
  const int i = rowB * 32 + lane;                      // my row box (i < KPAD)
  unsigned* mw = mask + ((size_t)img * KPAD + i) * NW + colB;
  if (colB < rowB) { *mw = 0u; return; }               // all j < i in this tile

  float4 rb = sobox[(size_t)img * KPAD + i];
  float areaA = (rb.z - rb.x) * (rb.w - rb.y);
  unsigned bits = 0u;
#pragma unroll 8
  for (int c = 0; c < 32; ++c) {
    float4 b = cb[c];
    float ix = fminf(rb.z, b.z) - fmaxf(rb.x, b.x);
    float iy = fminf(rb.w, b.w) - fmaxf(rb.y, b.y);
    float inter = fmaxf(ix, 0.f) * fmaxf(iy, 0.f);
    float areaB = (b.z - b.x) * (b.w - b.y);
    float iou = inter / (areaA + areaB - inter);
    int j = colB * 32 + c;
    if ((iou > NMS_T) && (j > i)) bits |= (1u << c);
  }
  *mw = bits;
}

// ---------------- K6: sequential greedy reduce (one wave per image) + output
__global__ __launch_bounds__(32) void nms_reduce_kernel(const unsigned* __restrict__ mask,
                                                        const unsigned* __restrict__ init_removed,
                                                        const float4* __restrict__ sbox,
                                                        const float* __restrict__ sscore,
                                                        float* __restrict__ out) {
  const int img = blockIdx.x, lane = threadIdx.x;
  __shared__ unsigned removed[NW];
  for (int w = lane; w < NW; w += 32) removed[w] = init_removed[img * NW + w];
  for (int z = lane; z < POST_N * 5; z += 32) out[(size_t)img * POST_N * 5 + z] = 0.f;
  __syncthreads();

  int cnt = 0;                                   // identical across all 32 lanes
  for (int i = 0; i < KTOT; ++i) {
    unsigned wword = removed[i >> 5];            // uniform LDS broadcast
    if (!((wword >> (i & 31)) & 1u)) {
      if (cnt < POST_N && lane < 5) {            // write one output row
        size_t base = (size_t)img * POST_N * 5 + (size_t)cnt * 5;
        size_t gi = (size_t)img * KPAD + i;
        float v;
        if      (lane == 0) v = sbox[gi].x;
        else if (lane == 1) v = sbox[gi].y;
        else if (lane == 2) v = sbox[gi].z;
        else if (lane == 3) v = sbox[gi].w;
        else                v = 1.f / (1.f + expf(-sscore[gi]));  // sigmoid
        out[base + lane] = v;
      }
      ++cnt;
      if (cnt >= POST_N) break;                  // further suppression can't affect output
      const unsigned* mr = mask + ((size_t)img * KPAD + i) * NW;
      for (int w = lane; w < NW; w += 32) removed[w] |= mr[w];   // row i suppresses
    }
    __syncthreads();
  }
}

// ---------------- host-side launch ----------------
extern "C" void kernel_launch(void* const* d_in, const int* in_sizes, int n_in,
                              void* d_out, int out_size, void* d_ws, size_t ws_size,
                              hipStream_t stream) {
  (void)in_sizes; (void)n_in; (void)out_size; (void)ws_size;
  const float* obj     = (const float*)d_in[0];   // [8, 159882]
  const float* deltas  = (const float*)d_in[1];   // [8, 159882, 4]
  const float* anchors = (const float*)d_in[2];   // [159882, 4]
  float* out = (float*)d_out;                     // [8, 1000, 5]

  char* w = (char*)d_ws;
  auto alloc = [&](size_t bytes) -> char* {
    char* p = w; w += (bytes + 255) & ~(size_t)255; return p;
  };
  float*    cand_score = (float*)   alloc(sizeof(float)    * NIMG * KTOT);
  int*      cand_aidx  = (int*)     alloc(sizeof(int)      * NIMG * KTOT);
  int*      cand_valid = (int*)     alloc(sizeof(int)      * NIMG * KTOT);
  float4*   cand_box   = (float4*)  alloc(sizeof(float4)   * NIMG * KTOT);
  float4*   cand_obox  = (float4*)  alloc(sizeof(float4)   * NIMG * KTOT);
  int*      perm       = (int*)     alloc(sizeof(int)      * NIMG * KTOT);
  float4*   sbox       = (float4*)  alloc(sizeof(float4)   * NIMG * KPAD);
  float4*   sobox      = (float4*)  alloc(sizeof(float4)   * NIMG * KPAD);
  float*    sscore     = (float*)   alloc(sizeof(float)    * NIMG * KPAD);
  unsigned* initrm     = (unsigned*)alloc(sizeof(unsigned) * NIMG * NW);
  unsigned* mask       = (unsigned*)alloc(sizeof(unsigned) * (size_t)NIMG * KPAD * NW); // ~20.4MB

  init_kernel<<<(NIMG * NW + 255) / 256, 256, 0, stream>>>(initrm);
  topk_kernel<<<dim3(NLEV, NIMG), 1024, 0, stream>>>(obj, cand_score, cand_aidx);
  decode_kernel<<<(NIMG * KTOT + 255) / 256, 256, 0, stream>>>(deltas, anchors, cand_aidx,
                                                               cand_box, cand_obox, cand_valid);
  sort_kernel<<<NIMG, 1024, 0, stream>>>(cand_score, cand_valid, perm);
  gather_kernel<<<(NIMG * KPAD + 255) / 256, 256, 0, stream>>>(perm, cand_box, cand_obox,
                                                               cand_score, cand_valid,
                                                               sbox, sobox, sscore, initrm);
  nms_mask_kernel<<<dim3(NRB, NW, NIMG), 256, 0, stream>>>(sobox, mask);
  nms_reduce_kernel<<<NIMG, 32, 0, stream>>>(mask, initrm, sbox, sscore, out);
}